// Deblock_68831145885827
// MI455X (gfx1250) — compile-verified
//
#include <hip/hip_runtime.h>

typedef __attribute__((ext_vector_type(2))) float v2f;
typedef __attribute__((ext_vector_type(8))) float v8f;

constexpr int BS = 64, N_COARSE = 2562, N_FINE = 10242, E_FINE = 61440;
constexpr int C_IN = 128, C_OUT = 64;
constexpr int R  = BS * N_COARSE;               // 163968 coarse rows (divisible by 16)
constexpr int NF = BS * N_FINE;                 // 655488 fine nodes
constexpr long long E = (long long)BS * E_FINE; // 3932160 directed edges

// ---- init: zero output accumulator, deg=1 (self loop), inv=-1 ----
__global__ void init_kernel(float* __restrict__ out, float* __restrict__ deg,
                            int* __restrict__ inv) {
  long long i = (long long)blockIdx.x * blockDim.x + threadIdx.x;
  long long total = (long long)NF * C_OUT;
  if (i < total) out[i] = 0.0f;
  if (i < NF) { deg[i] = 1.0f; inv[i] = -1; }
}

// ---- inverse unpool map: fine node -> coarse row index ----
__global__ void build_inv(const long long* __restrict__ mask, int* __restrict__ inv) {
  int i = blockIdx.x * blockDim.x + threadIdx.x;
  if (i < R) inv[(int)mask[i]] = i;
}

// ---- degree accumulation over dst ----
__global__ void deg_kernel(const int* __restrict__ dst, float* __restrict__ deg) {
  long long e = (long long)blockIdx.x * blockDim.x + threadIdx.x;
  if (e < E) atomicAdd(deg + dst[e], 1.0f);
}

// ---- deg -> rsqrt(deg) in place ----
__global__ void norm_kernel(float* __restrict__ deg) {
  int i = blockIdx.x * blockDim.x + threadIdx.x;
  if (i < NF) deg[i] = rsqrtf(deg[i]);
}

// ---- compacted GEMM hc[R,64] = x[R,128] @ W[128,64] via V_WMMA_F32_16X16X4_F32 ----
// Block = 128 threads = 4 waves; wave w owns column tile [16w,16w+16);
// grid.x = R/16 row tiles. EXEC always all-1s (exact multiples).
__global__ void gemm_wmma(const float* __restrict__ x, const float* __restrict__ W,
                          float* __restrict__ hc) {
  const int wave = threadIdx.x >> 5;     // 0..3 -> column tile
  const int lane = threadIdx.x & 31;
  const int half = lane >> 4;            // K-pair select per ISA A/B layout
  const int l16  = lane & 15;
  const int rowBase = blockIdx.x * 16;
  const int colBase = wave * 16;

  // A frag: lane(l16,half) holds A[M=l16][K = k+2*half .. +1]
  const float* xr = x + (long long)(rowBase + l16) * C_IN + 2 * half;
  // B frag: lane holds B[K][N=colBase+l16] for the same K pair
  const float* wc = W + colBase + l16;

  v8f acc = {};
#pragma unroll
  for (int k = 0; k < C_IN; k += 4) {
    v2f a, b;
    a.x = xr[k];
    a.y = xr[k + 1];
    b.x = wc[(long long)(k + 2 * half + 0) * C_OUT];
    b.y = wc[(long long)(k + 2 * half + 1) * C_OUT];
    acc = __builtin_amdgcn_wmma_f32_16x16x4_f32(
        /*neg_a=*/false, a, /*neg_b=*/false, b,
        /*c_mod=*/(short)0, acc, /*reuse_a=*/false, /*reuse_b=*/false);
  }

  // D layout: VGPR i -> M = 8*half + i, N = l16
#pragma unroll
  for (int i = 0; i < 8; ++i) {
    int m = half * 8 + i;
    hc[(long long)(rowBase + m) * C_OUT + colBase + l16] = acc[i];
  }
}

// ---- edge aggregation: 16 lanes/edge, float4 channels, skip zero-h sources ----
__global__ void edge_agg(const int* __restrict__ src, const int* __restrict__ dst,
                         const int* __restrict__ inv, const float* __restrict__ nrm,
                         const float* __restrict__ hc, float* __restrict__ out) {
  long long t = (long long)blockIdx.x * blockDim.x + threadIdx.x;
  long long e = t >> 4;           // 16 threads per edge
  int c = ((int)t & 15) * 4;      // 4 channels per thread
  if (e >= E) return;
  int s = src[e];
  int ci = inv[s];
  if (ci < 0) return;             // h[src] == 0 -> no contribution (~75% of edges)
  int d = dst[e];
  float coef = nrm[s] * nrm[d];
  const float4 hv = *(const float4*)(hc + (long long)ci * C_OUT + c);
  float* o = out + (long long)d * C_OUT + c;
  atomicAdd(o + 0, hv.x * coef);
  atomicAdd(o + 1, hv.y * coef);
  atomicAdd(o + 2, hv.z * coef);
  atomicAdd(o + 3, hv.w * coef);
}

// ---- self-loop + bias + ELU, in place on the accumulator ----
__global__ void finalize(float* __restrict__ out, const float* __restrict__ hc,
                         const int* __restrict__ inv, const float* __restrict__ nrm,
                         const float* __restrict__ bias) {
  long long i = (long long)blockIdx.x * blockDim.x + threadIdx.x;
  if (i >= (long long)NF * C_OUT) return;
  int n = (int)(i >> 6);
  int c = (int)(i & 63);
  float v = out[i];
  int ci = inv[n];
  if (ci >= 0) {
    float nn = nrm[n];
    v += hc[(long long)ci * C_OUT + c] * (nn * nn);
  }
  v += bias[c];
  out[i] = (v > 0.0f) ? v : (expf(v) - 1.0f);
}

extern "C" void kernel_launch(void* const* d_in, const int* in_sizes, int n_in,
                              void* d_out, int out_size, void* d_ws, size_t ws_size,
                              hipStream_t stream) {
  (void)in_sizes; (void)n_in; (void)out_size; (void)ws_size;
  const float*      x    = (const float*)d_in[0];
  const float*      W    = (const float*)d_in[1];
  const float*      bias = (const float*)d_in[2];
  const int*        ei   = (const int*)d_in[3];
  const long long*  mask = (const long long*)d_in[4];
  const int* src = ei;
  const int* dst = ei + E;
  float* out = (float*)d_out;

  // workspace: hc (42 MB, L2-resident) | norm (2.6 MB) | inv (2.6 MB)
  char* ws = (char*)d_ws;
  float* hc = (float*)ws;
  ws += (((size_t)R * C_OUT * sizeof(float)) + 255) & ~(size_t)255;
  float* nrm = (float*)ws;
  ws += (((size_t)NF * sizeof(float)) + 255) & ~(size_t)255;
  int* inv = (int*)ws;

  const long long totalOut = (long long)NF * C_OUT;
  init_kernel<<<(int)((totalOut + 255) / 256), 256, 0, stream>>>(out, nrm, inv);
  build_inv<<<(R + 255) / 256, 256, 0, stream>>>(mask, inv);
  deg_kernel<<<(int)((E + 255) / 256), 256, 0, stream>>>(dst, nrm);
  norm_kernel<<<(NF + 255) / 256, 256, 0, stream>>>(nrm);
  gemm_wmma<<<R / 16, 128, 0, stream>>>(x, W, hc);
  const long long aggThreads = E * 16;
  edge_agg<<<(int)((aggThreads + 255) / 256), 256, 0, stream>>>(src, dst, inv, nrm, hc, out);
  finalize<<<(int)((totalOut + 255) / 256), 256, 0, stream>>>(out, hc, inv, nrm, bias);
}